// SAGENC_82832739270702
// MI455X (gfx1250) — compile-verified
//
#include <hip/hip_runtime.h>
#include <hip/hip_bf16.h>

// ---------------------------------------------------------------------------
// CDNA5 (gfx1250) WMMA fp32 path: V_WMMA_F32_16X16X4_F32, wave32.
// ---------------------------------------------------------------------------
typedef float v2f __attribute__((ext_vector_type(2)));
typedef float v8f __attribute__((ext_vector_type(8)));

__device__ __forceinline__ v8f wmma_f32(v2f a, v2f b, v8f c) {
  // 8 args: (neg_a, A, neg_b, B, c_mod, C, reuse_a, reuse_b)
  return __builtin_amdgcn_wmma_f32_16x16x4_f32(false, a, false, b, (short)0, c, false, false);
}

// ---------------------------------------------------------------------------
// fp32 GEMM, M%32==0 && Nout%32==0. One wave owns a 32x32 output macro-tile
// (2x2 WMMA tiles): per k-step 4 A-loads + 4 B-loads feed 4 independent WMMAs.
// Compile-time (K, Nout) -> all fragment addresses become base + immediate
// offsets (A offs <= 2KB, W offs <= 512KB, both within signed 24-bit IOFFSET).
// A-frag (16x4): lanes 0-15 rows, {k0,k0+1}; lanes 16-31 rows, {k0+2,k0+3}.
// C/D: VGPR r -> M=r (lanes 0-15) / M=r+8 (lanes 16-31), N = lane&15.
// ---------------------------------------------------------------------------
template <int K, int Nout>
__global__ void gemm32_tmpl(const float* __restrict__ A,
                            const float* __restrict__ W,
                            const float* __restrict__ bias,
                            float* __restrict__ C,
                            int M, int doRelu) {
  const int lane = threadIdx.x & 31;
  const int wave = threadIdx.x >> 5;
  const int mtiles = M >> 5;
  constexpr int ntiles = Nout >> 5;
  const long total = (long)mtiles * ntiles;
  const long t = (long)blockIdx.x * (blockDim.x >> 5) + wave;
  if (t >= total) return;                    // uniform per wave -> EXEC all-ones
  const int mt = (int)(t / ntiles);
  const int nt = (int)(t % ntiles);
  const int half = lane >> 4;
  const int l16  = lane & 15;
  const int row0 = mt * 32 + l16;
  const int col0 = nt * 32 + l16;
  const int koff = half << 1;                // lane-half k offset {0,2}

  const float* a0p = A + (long)row0 * K + koff;   // [kt*4] , [kt*4+1]
  const float* a1p = a0p + (long)16 * K;
  const float* wp  = W + (long)koff * Nout + col0; // [kt*4*Nout (+Nout)(+16)]

  v8f acc00 = {}, acc01 = {}, acc10 = {}, acc11 = {};
  constexpr int kfull = K >> 2;

#pragma unroll 8
  for (int kt = 0; kt < kfull; ++kt) {
    const long wo = (long)kt * 4 * Nout;
    v2f a0 = {a0p[kt * 4], a0p[kt * 4 + 1]};
    v2f a1 = {a1p[kt * 4], a1p[kt * 4 + 1]};
    v2f b0 = {wp[wo],      wp[wo + Nout]};
    v2f b1 = {wp[wo + 16], wp[wo + Nout + 16]};
    acc00 = wmma_f32(a0, b0, acc00);
    acc01 = wmma_f32(a0, b1, acc01);
    acc10 = wmma_f32(a1, b0, acc10);
    acc11 = wmma_f32(a1, b1, acc11);
  }

  if constexpr ((K & 3) != 0) {              // K tail, one guarded step
    const int kb = kfull << 2;               // lane k = kb + koff {+1}
    const int k0 = kb + koff;
    const long wo = (long)kb * Nout;
    v2f a0, a1, b0, b1;
    a0.x = (k0     < K) ? a0p[kb]     : 0.0f;
    a0.y = (k0 + 1 < K) ? a0p[kb + 1] : 0.0f;
    a1.x = (k0     < K) ? a1p[kb]     : 0.0f;
    a1.y = (k0 + 1 < K) ? a1p[kb + 1] : 0.0f;
    b0.x = (k0     < K) ? wp[wo]             : 0.0f;
    b0.y = (k0 + 1 < K) ? wp[wo + Nout]      : 0.0f;
    b1.x = (k0     < K) ? wp[wo + 16]        : 0.0f;
    b1.y = (k0 + 1 < K) ? wp[wo + Nout + 16] : 0.0f;
    acc00 = wmma_f32(a0, b0, acc00);
    acc01 = wmma_f32(a0, b1, acc01);
    acc10 = wmma_f32(a1, b0, acc10);
    acc11 = wmma_f32(a1, b1, acc11);
  }

  const float bv0 = bias ? bias[col0]      : 0.0f;
  const float bv1 = bias ? bias[col0 + 16] : 0.0f;
  float* c0 = C + (long)(mt * 32 + half * 8) * Nout + col0;
#pragma unroll
  for (int r = 0; r < 8; ++r) {
    float v00 = acc00[r] + bv0, v01 = acc01[r] + bv1;
    float v10 = acc10[r] + bv0, v11 = acc11[r] + bv1;
    if (doRelu) {
      v00 = fmaxf(v00, 0.0f); v01 = fmaxf(v01, 0.0f);
      v10 = fmaxf(v10, 0.0f); v11 = fmaxf(v11, 0.0f);
    }
    c0[(long)r * Nout]              = v00;
    c0[(long)r * Nout + 16]         = v01;
    c0[(long)(r + 16) * Nout]       = v10;
    c0[(long)(r + 16) * Nout + 16]  = v11;
  }
}

// ---------------------------------------------------------------------------
// Runtime-shape fallback (software-pipelined), same layout contracts.
// ---------------------------------------------------------------------------
__device__ __forceinline__ void load_frags(const float* __restrict__ A0,
                                           const float* __restrict__ A1,
                                           const float* __restrict__ W,
                                           int Nout, int col0, int col1, int k0,
                                           v2f& a0, v2f& a1, v2f& b0, v2f& b1) {
  a0.x = A0[k0]; a0.y = A0[k0 + 1];
  a1.x = A1[k0]; a1.y = A1[k0 + 1];
  const float* Wk0 = W + (long)k0 * Nout;
  const float* Wk1 = Wk0 + Nout;
  b0.x = Wk0[col0]; b0.y = Wk1[col0];
  b1.x = Wk0[col1]; b1.y = Wk1[col1];
}

__global__ void gemm32_generic(const float* __restrict__ A,
                               const float* __restrict__ W,
                               const float* __restrict__ bias,
                               float* __restrict__ C,
                               int M, int K, int Nout, int doRelu) {
  const int lane = threadIdx.x & 31;
  const int wave = threadIdx.x >> 5;
  const int mtiles = M >> 5;
  const int ntiles = Nout >> 5;
  const long total = (long)mtiles * ntiles;
  const long t = (long)blockIdx.x * (blockDim.x >> 5) + wave;
  if (t >= total) return;
  const int mt = (int)(t / ntiles);
  const int nt = (int)(t % ntiles);
  const int half = lane >> 4;
  const int l16  = lane & 15;
  const int row0 = mt * 32 + l16;
  const int col0 = nt * 32 + l16;
  const int col1 = col0 + 16;
  const float* A0 = A + (long)row0 * K;
  const float* A1 = A0 + (long)16 * K;
  const int koff = half << 1;

  v8f acc00 = {}, acc01 = {}, acc10 = {}, acc11 = {};
  const int kfull = K >> 2;

  if (kfull > 0) {
    v2f a0, a1, b0, b1;
    load_frags(A0, A1, W, Nout, col0, col1, koff, a0, a1, b0, b1);
    for (int kt = 0; kt < kfull - 1; ++kt) {
      v2f na0, na1, nb0, nb1;
      load_frags(A0, A1, W, Nout, col0, col1, ((kt + 1) << 2) + koff,
                 na0, na1, nb0, nb1);
      acc00 = wmma_f32(a0, b0, acc00);
      acc01 = wmma_f32(a0, b1, acc01);
      acc10 = wmma_f32(a1, b0, acc10);
      acc11 = wmma_f32(a1, b1, acc11);
      a0 = na0; a1 = na1; b0 = nb0; b1 = nb1;
    }
    acc00 = wmma_f32(a0, b0, acc00);
    acc01 = wmma_f32(a0, b1, acc01);
    acc10 = wmma_f32(a1, b0, acc10);
    acc11 = wmma_f32(a1, b1, acc11);
  }

  if (K & 3) {
    const int k0 = (kfull << 2) + koff;
    v2f a0, a1, b0, b1;
    a0.x = (k0     < K) ? A0[k0]     : 0.0f;
    a0.y = (k0 + 1 < K) ? A0[k0 + 1] : 0.0f;
    a1.x = (k0     < K) ? A1[k0]     : 0.0f;
    a1.y = (k0 + 1 < K) ? A1[k0 + 1] : 0.0f;
    b0.x = (k0     < K) ? W[(long)(k0)     * Nout + col0] : 0.0f;
    b0.y = (k0 + 1 < K) ? W[(long)(k0 + 1) * Nout + col0] : 0.0f;
    b1.x = (k0     < K) ? W[(long)(k0)     * Nout + col1] : 0.0f;
    b1.y = (k0 + 1 < K) ? W[(long)(k0 + 1) * Nout + col1] : 0.0f;
    acc00 = wmma_f32(a0, b0, acc00);
    acc01 = wmma_f32(a0, b1, acc01);
    acc10 = wmma_f32(a1, b0, acc10);
    acc11 = wmma_f32(a1, b1, acc11);
  }

  const float bv0 = bias ? bias[col0] : 0.0f;
  const float bv1 = bias ? bias[col1] : 0.0f;
#pragma unroll
  for (int r = 0; r < 8; ++r) {
    const int rowA = mt * 32 + r + half * 8;
    const int rowB = rowA + 16;
    float v00 = acc00[r] + bv0, v01 = acc01[r] + bv1;
    float v10 = acc10[r] + bv0, v11 = acc11[r] + bv1;
    if (doRelu) {
      v00 = fmaxf(v00, 0.0f); v01 = fmaxf(v01, 0.0f);
      v10 = fmaxf(v10, 0.0f); v11 = fmaxf(v11, 0.0f);
    }
    C[(long)rowA * Nout + col0] = v00;
    C[(long)rowA * Nout + col1] = v01;
    C[(long)rowB * Nout + col0] = v10;
    C[(long)rowB * Nout + col1] = v11;
  }
}

// ---------------------------------------------------------------------------
// Final layer: out[m] = A[m,:]·w + b   (Nout==1)  — wave per row.
// ---------------------------------------------------------------------------
__global__ void matvec_kernel(const float* __restrict__ A,
                              const float* __restrict__ w,
                              const float* __restrict__ b,
                              float* __restrict__ out, int M, int K) {
  const int lane = threadIdx.x & 31;
  const long row = ((long)blockIdx.x * blockDim.x + threadIdx.x) >> 5;
  if (row >= M) return;
  const float* a = A + row * (long)K;
  float s = 0.0f;
  for (int k = lane; k < K; k += 32) s += a[k] * w[k];
#pragma unroll
  for (int m = 16; m; m >>= 1) s += __shfl_xor(s, m, 32);
  if (lane == 0) out[row] = s + b[0];
}

// ---------------------------------------------------------------------------
// Attention scores per node: a_s[n] = hp[n,:]·att_src, a_d[n] = hp[n,:]·att_dst
// ---------------------------------------------------------------------------
__global__ void gat_attn_kernel(const float* __restrict__ hp,
                                const float* __restrict__ att_src,
                                const float* __restrict__ att_dst,
                                float* __restrict__ a_s, float* __restrict__ a_d,
                                int N, int D) {
  const int lane = threadIdx.x & 31;
  const long node = ((long)blockIdx.x * blockDim.x + threadIdx.x) >> 5;
  if (node >= N) return;
  const float* row = hp + node * (long)D;
  float s = 0.0f, d = 0.0f;
  for (int f = lane; f < D; f += 32) {
    const float h = row[f];
    s += h * att_src[f];
    d += h * att_dst[f];
  }
#pragma unroll
  for (int m = 16; m; m >>= 1) { s += __shfl_xor(s, m, 32); d += __shfl_xor(d, m, 32); }
  if (lane == 0) { a_s[node] = s; a_d[node] = d; }
}

// ---- monotonic float<->int key (involution) for atomicMax on floats --------
__device__ __forceinline__ int   fkey(float f) { int i = __float_as_int(f); return i >= 0 ? i : (i ^ 0x7fffffff); }
__device__ __forceinline__ float fkeyinv(int k) { return __int_as_float(k >= 0 ? k : (k ^ 0x7fffffff)); }

__device__ __forceinline__ void edge_sd(const int* __restrict__ ei, int E, long i,
                                        int& s, int& d) {
  if (i < E) { s = ei[i]; d = ei[(long)E + i]; } else { s = d = (int)(i - E); }
}

// ---------------------------------------------------------------------------
// Edge pass 1: leaky_relu score, per-dst segment max (atomicMax on key), degree
// ---------------------------------------------------------------------------
__global__ void edge_score_kernel(const int* __restrict__ ei,
                                  const float* __restrict__ a_s,
                                  const float* __restrict__ a_d,
                                  float* __restrict__ scores,
                                  int* __restrict__ emaxKey,
                                  float* __restrict__ deg,
                                  int E, int N) {
  const long i = (long)blockIdx.x * blockDim.x + threadIdx.x;
  const long tot = (long)E + N;
  if (i >= tot) return;
  int s, d; edge_sd(ei, E, i, s, d);
  float e = a_s[s] + a_d[d];
  e = (e > 0.0f) ? e : 0.2f * e;             // leaky_relu, slope 0.2
  scores[i] = e;
  atomicMax(&emaxKey[d], fkey(e));
  atomicAdd(&deg[d], 1.0f);
}

// ---------------------------------------------------------------------------
// Edge pass 2: ee = exp(e - emax[dst]); denom[dst] += ee
// ---------------------------------------------------------------------------
__global__ void edge_exp_kernel(const int* __restrict__ ei,
                                const int* __restrict__ emaxKey,
                                float* __restrict__ scores,
                                float* __restrict__ denom,
                                int E, int N) {
  const long i = (long)blockIdx.x * blockDim.x + threadIdx.x;
  const long tot = (long)E + N;
  if (i >= tot) return;
  int s, d; edge_sd(ei, E, i, s, d);
  const float ee = __expf(scores[i] - fkeyinv(emaxKey[d]));
  scores[i] = ee;
  atomicAdd(&denom[d], ee);
}

// ---------------------------------------------------------------------------
// GAT scatter: h_out[dst,:] += alpha * hp[src,:]   (wave per edge)
// ---------------------------------------------------------------------------
__global__ void gat_scatter_kernel(const int* __restrict__ ei,
                                   const float* __restrict__ scores,
                                   const float* __restrict__ denom,
                                   const float* __restrict__ hp,
                                   float* __restrict__ hout,
                                   int E, int N, int D) {
  const int lane = threadIdx.x & 31;
  const long e = ((long)blockIdx.x * blockDim.x + threadIdx.x) >> 5;
  const long tot = (long)E + N;
  if (e >= tot) return;
  int s, d; edge_sd(ei, E, e, s, d);
  const float alpha = scores[e] / denom[d];
  const float* hs = hp + (long)s * D;
  float* hd = hout + (long)d * D;
  for (int f = lane; f < D; f += 32) atomicAdd(&hd[f], alpha * hs[f]);
}

// ---------------------------------------------------------------------------
// GCN scatter: h_out[dst,:] += d_inv[src]*d_inv[dst]*hw[src,:]
// ---------------------------------------------------------------------------
__global__ void gcn_scatter_kernel(const int* __restrict__ ei,
                                   const float* __restrict__ dinv,
                                   const float* __restrict__ hw,
                                   float* __restrict__ hout,
                                   int E, int N, int D) {
  const int lane = threadIdx.x & 31;
  const long e = ((long)blockIdx.x * blockDim.x + threadIdx.x) >> 5;
  const long tot = (long)E + N;
  if (e >= tot) return;
  int s, d; edge_sd(ei, E, e, s, d);
  const float norm = dinv[s] * dinv[d];
  const float* hs = hw + (long)s * D;
  float* hd = hout + (long)d * D;
  for (int f = lane; f < D; f += 32) atomicAdd(&hd[f], norm * hs[f]);
}

// ---- small elementwise kernels --------------------------------------------
__global__ void fill_f32(float* p, float v, long n) {
  const long i = (long)blockIdx.x * blockDim.x + threadIdx.x;
  if (i < n) p[i] = v;
}
__global__ void fill_i32(int* p, int v, long n) {
  const long i = (long)blockIdx.x * blockDim.x + threadIdx.x;
  if (i < n) p[i] = v;
}
__global__ void bias_relu_kernel(float* __restrict__ h, const float* __restrict__ b,
                                 long n, int D) {
  const long i = (long)blockIdx.x * blockDim.x + threadIdx.x;
  if (i >= n) return;
  const int f = (int)(i - (i / D) * D);
  h[i] = fmaxf(h[i] + b[f], 0.0f);
}
__global__ void dinv_kernel(float* __restrict__ deg, int N) {
  const long i = (long)blockIdx.x * blockDim.x + threadIdx.x;
  if (i >= N) return;
  const float d = deg[i];
  deg[i] = (d > 0.0f) ? rsqrtf(d) : 0.0f;
}
// bias + relu + segment-max pool (values >= 0, g pre-filled with 0 bits)
__global__ void gcn_finalize_pool_kernel(const float* __restrict__ h,
                                         const float* __restrict__ b,
                                         const int* __restrict__ batch,
                                         float* __restrict__ g,
                                         long n, int D) {
  const long i = (long)blockIdx.x * blockDim.x + threadIdx.x;
  if (i >= n) return;
  const long node = i / D;
  const int f = (int)(i - node * D);
  const float v = fmaxf(h[i] + b[f], 0.0f);
  atomicMax((int*)&g[(long)batch[node] * D + f], __float_as_int(v));
}

// ---------------------------------------------------------------------------
// Host side: shape-specialized dispatch
// ---------------------------------------------------------------------------
static inline void launch_gemm(const float* A, const float* W, const float* bias,
                               float* C, int M, int K, int Nout, int relu,
                               hipStream_t s) {
  const long tiles = (long)(M >> 5) * (Nout >> 5);   // 32x32 macro tiles
  const int blocks = (int)((tiles + 7) / 8);         // 8 waves per 256-thread block
#define GEMM_CASE(KK, NN)                                                     \
  if (K == KK && Nout == NN) {                                                \
    gemm32_tmpl<KK, NN><<<blocks, 256, 0, s>>>(A, W, bias, C, M, relu);       \
    return;                                                                   \
  }
  GEMM_CASE(53, 128)    // x @ w1
  GEMM_CASE(128, 64)    // h1 @ w2
  GEMM_CASE(64, 64)     // h2 @ gat_w
  GEMM_CASE(64, 128)    // h @ gcn_w
  GEMM_CASE(128, 512)   // g @ fc1
  GEMM_CASE(512, 256)   // z1 @ fc2
#undef GEMM_CASE
  gemm32_generic<<<blocks, 256, 0, s>>>(A, W, bias, C, M, K, Nout, relu);
}

extern "C" void kernel_launch(void* const* d_in, const int* in_sizes, int n_in,
                              void* d_out, int out_size, void* d_ws, size_t ws_size,
                              hipStream_t stream) {
  // ---- inputs (setup_inputs order) ----
  const float* x        = (const float*)d_in[0];
  const int*   ei       = (const int*)d_in[1];   // edge_inter [2,E]
  /* edge_intra d_in[2] unused by reference */
  const int*   batch    = (const int*)d_in[3];
  /* num_graphs d_in[4]: G == out_size */
  const float* w1       = (const float*)d_in[5];
  const float* b1       = (const float*)d_in[6];
  const float* w2       = (const float*)d_in[7];
  const float* b2       = (const float*)d_in[8];
  const float* gat_w    = (const float*)d_in[9];
  const float* att_src  = (const float*)d_in[10];
  const float* att_dst  = (const float*)d_in[11];
  const float* gat_b    = (const float*)d_in[12];
  const float* gcn_w    = (const float*)d_in[13];
  const float* gcn_b    = (const float*)d_in[14];
  const float* fc1_w    = (const float*)d_in[15];
  const float* fc1_b    = (const float*)d_in[16];
  const float* fc2_w    = (const float*)d_in[17];
  const float* fc2_b    = (const float*)d_in[18];
  const float* out_w    = (const float*)d_in[19];
  const float* out_b    = (const float*)d_in[20];
  float* out = (float*)d_out;

  const int N    = in_sizes[3];            // 100000 (mult of 32)
  const int E    = in_sizes[1] / 2;        // 1600000
  const int F_IN = in_sizes[0] / N;        // 53
  const int D1   = in_sizes[6];            // 128
  const int D2   = in_sizes[8];            // 64
  const int DG   = in_sizes[14];           // 2*D2 = 128
  const int H1   = in_sizes[16];           // 512
  const int H2   = in_sizes[18];           // 256
  const int G    = out_size;               // 512 (mult of 32)
  const long EN  = (long)E + N;

  // ---- workspace carve-up (256B aligned) ----
  char* base = (char*)d_ws;
  size_t off = 0;
  auto alloc = [&](size_t bytes) -> char* {
    char* p = base + off;
    off = (off + bytes + 255) & ~(size_t)255;
    return p;
  };
  float* bufD    = (float*)alloc((size_t)N * D1 * 4);  // h1, later GCN accumulator
  float* bufA    = (float*)alloc((size_t)N * D2 * 4);  // h2, later GAT accumulator
  float* bufB    = (float*)alloc((size_t)N * D2 * 4);  // hp
  float* bufC    = (float*)alloc((size_t)N * DG * 4);  // hw
  float* scores  = (float*)alloc((size_t)EN * 4);
  float* a_s     = (float*)alloc((size_t)N * 4);
  float* a_d     = (float*)alloc((size_t)N * 4);
  int*   emaxKey = (int*)  alloc((size_t)N * 4);
  float* denom   = (float*)alloc((size_t)N * 4);
  float* deg     = (float*)alloc((size_t)N * 4);       // becomes d_inv in place
  float* gpool   = (float*)alloc((size_t)G * DG * 4);
  float* z1      = (float*)alloc((size_t)G * H1 * 4);
  float* z2      = (float*)alloc((size_t)G * H2 * 4);
  (void)ws_size; (void)n_in;

  auto gridOf = [](long n) { return (unsigned)((n + 255) / 256); };

  // 1-2) node embed: h1 = relu(x@w1+b1); h2 = relu(h1@w2+b2)
  launch_gemm(x,    w1, b1, bufD, N, F_IN, D1, 1, stream);
  launch_gemm(bufD, w2, b2, bufA, N, D1,   D2, 1, stream);

  // 3) hp = h2 @ gat_w
  launch_gemm(bufA, gat_w, nullptr, bufB, N, D2, D2, 0, stream);

  // 4) per-node attention coefficients
  gat_attn_kernel<<<gridOf((long)N * 32), 256, 0, stream>>>(bufB, att_src, att_dst,
                                                            a_s, a_d, N, D2);

  // 5) init segment buffers + GAT accumulator (h2 is dead now)
  fill_i32<<<gridOf(N), 256, 0, stream>>>(emaxKey, (int)0x80000000, N);
  fill_f32<<<gridOf(N), 256, 0, stream>>>(denom, 0.0f, N);
  fill_f32<<<gridOf(N), 256, 0, stream>>>(deg,   0.0f, N);
  fill_f32<<<gridOf((long)N * D2), 256, 0, stream>>>(bufA, 0.0f, (long)N * D2);

  // 6-8) edge softmax + weighted scatter
  edge_score_kernel<<<gridOf(EN), 256, 0, stream>>>(ei, a_s, a_d, scores, emaxKey,
                                                    deg, E, N);
  edge_exp_kernel<<<gridOf(EN), 256, 0, stream>>>(ei, emaxKey, scores, denom, E, N);
  gat_scatter_kernel<<<gridOf(EN * 32), 256, 0, stream>>>(ei, scores, denom, bufB,
                                                          bufA, E, N, D2);

  // 9-10) GAT epilogue + d_inv
  bias_relu_kernel<<<gridOf((long)N * D2), 256, 0, stream>>>(bufA, gat_b,
                                                             (long)N * D2, D2);
  dinv_kernel<<<gridOf(N), 256, 0, stream>>>(deg, N);

  // 11-13) GCN: hw = h @ gcn_w; scatter with symmetric norm
  launch_gemm(bufA, gcn_w, nullptr, bufC, N, D2, DG, 0, stream);
  fill_f32<<<gridOf((long)N * DG), 256, 0, stream>>>(bufD, 0.0f, (long)N * DG);
  gcn_scatter_kernel<<<gridOf(EN * 32), 256, 0, stream>>>(ei, deg, bufC, bufD,
                                                          E, N, DG);

  // 14-15) GCN epilogue + global max pool
  fill_f32<<<gridOf((long)G * DG), 256, 0, stream>>>(gpool, 0.0f, (long)G * DG);
  gcn_finalize_pool_kernel<<<gridOf((long)N * DG), 256, 0, stream>>>(
      bufD, gcn_b, batch, gpool, (long)N * DG, DG);

  // 16-18) MLP head
  launch_gemm(gpool, fc1_w, fc1_b, z1,  G, DG, H1, 1, stream);
  launch_gemm(z1,    fc2_w, fc2_b, z2,  G, H1, H2, 1, stream);
  matvec_kernel<<<gridOf((long)G * 32), 256, 0, stream>>>(z2, out_w, out_b, out, G, H2);
}